// TemporalScaledDotProductforCrossAttention_6803228197041
// MI455X (gfx1250) — compile-verified
//
#include <hip/hip_runtime.h>

typedef __attribute__((ext_vector_type(2))) float v2f;
typedef __attribute__((ext_vector_type(8))) float v8f;

#define NEG_INF_F (-1e9f)

// D = A(16x4,f32) * B(4x16,f32) + C(16x16,f32), full-precision CDNA5 WMMA.
__device__ __forceinline__ v8f wmma4(v2f a, v2f b, v8f c) {
  return __builtin_amdgcn_wmma_f32_16x16x4_f32(
      /*neg_a=*/false, a, /*neg_b=*/false, b,
      /*c_mod=*/(short)0, c, /*reuse_a=*/false, /*reuse_b=*/false);
}

__global__ __launch_bounds__(256) void
fused_temporal_cross_attn(const float* __restrict__ Qf, const float* __restrict__ Kf,
                          const float* __restrict__ Vf, const float* __restrict__ Qs,
                          const float* __restrict__ Ks, const float* __restrict__ Vs,
                          const float* __restrict__ Kj, const float* __restrict__ Vfp,
                          const int* __restrict__ mask, float* __restrict__ out,
                          int n_nodes) {
  __shared__ int   smask[144];          // 12x12 mask
  __shared__ float zbuf[8][16 * 20];    // per-wave z staging, stride 20 (bank-spread)

  const int tid = threadIdx.x;
  if (tid < 144) smask[tid] = mask[tid];

  const int w    = tid >> 5;        // wave in block
  const int lane = tid & 31;
  const int r    = lane & 15;       // row index (M for A-op, N for B-op, col for C)
  const int rc   = (r < 12) ? r : 11;  // clamped row: pad rows load valid garbage
  const int half = lane >> 4;
  const int dsel = 2 * half;

  int  nid    = blockIdx.x * 8 + w;
  bool active = (nid < n_nodes);
  if (!active) nid = 0;             // keep loads valid, barriers uniform

  const size_t base = (size_t)nid * 768;   // 12*64 fp32 tile
  const float* qf = Qf + base; const float* kf = Kf + base;
  const float* vf = Vf + base; const float* qs = Qs + base;
  const float* ks = Ks + base; const float* vs = Vs + base;

  // FlowSpeed per-time-row constants (garbage for pad rows -> masked later).
  const float kj  = Kj[rc];
  const float ivf = 1.0f / (Vfp[rc] + 1e-5f);

  const float* Atab[4] = { qf, kf, ks, qs };
  const float* Btab[4] = { kf, qs, qf, ks };   // f==1 applies FlowSpeed to B
  const float* Vtab[4] = { vf, vf, vs, vs };

  float* zb = &zbuf[w][0];
  const size_t fstride = (size_t)n_nodes * 768;

  __syncthreads();  // smask visible

  // keep[p]: this lane's score entry (q=p+8*half, k=r) survives masking.
  // Lane-invariant across flavors; hoisted out of the flavor loop.
  bool keep[8];
#pragma unroll
  for (int p = 0; p < 8; ++p) {
    const int  q   = p + 8 * half;
    const bool inb = (r < 12) && (q < 12);
    const int  mi  = ((q < 12) ? q : 11) * 12 + rc;   // always in-bounds
    keep[p] = inb && (smask[mi] == 0);
  }

#pragma unroll
  for (int f = 0; f < 4; ++f) {
    // -------- scores: preload 16 A-chunks + 16 B-chunks, then 16 WMMAs ------
    const float* pA = Atab[f] + rc * 64 + dsel;
    const float* pB = Btab[f] + rc * 64 + dsel;
    v2f av[16], bv[16];
#pragma unroll
    for (int c = 0; c < 16; ++c) av[c] = *(const v2f*)(pA + 4 * c);
#pragma unroll
    for (int c = 0; c < 16; ++c) bv[c] = *(const v2f*)(pB + 4 * c);
    if (f == 1) {  // Qf_from_speed = Kj * (Qs - Qs^2 / (Vf_param + 1e-5))
#pragma unroll
      for (int c = 0; c < 16; ++c) {
        bv[c].x = kj * (bv[c].x - bv[c].x * bv[c].x * ivf);
        bv[c].y = kj * (bv[c].y - bv[c].y * bv[c].y * ivf);
      }
    }
    v8f acc = {};
#pragma unroll
    for (int c = 0; c < 16; ++c) acc = wmma4(av[c], bv[c], acc);

    // -------- scale + mask + row softmax (row q lives in 16 lanes) ----------
    float z[8];
#pragma unroll
    for (int p = 0; p < 8; ++p) {
      float s = keep[p] ? (acc[p] * 0.125f) : NEG_INF_F;  // 1/sqrt(64)
      float m = s;
#pragma unroll
      for (int off = 8; off >= 1; off >>= 1)
        m = fmaxf(m, __shfl_xor(m, off, 16));
      float e = __expf(s - m);
      e = (r < 12) ? e : 0.0f;                 // pad cols contribute nothing
      float sum = e;
#pragma unroll
      for (int off = 8; off >= 1; off >>= 1)
        sum += __shfl_xor(sum, off, 16);
      z[p] = e / sum;
    }

    // -------- bounce z through LDS into WMMA A-layout -----------------------
    __syncthreads();  // WAR: previous flavor's context reads finished
#pragma unroll
    for (int p = 0; p < 8; ++p)
      zb[(p + 8 * half) * 20 + r] = z[p];
    __syncthreads();  // RAW: z visible transposed

    // z A-chunks (K=12 -> 3 chunks), reused across all 4 N-tiles.
    v2f az[3];
#pragma unroll
    for (int cc = 0; cc < 3; ++cc) {
      const float* zp = zb + rc * 20 + 4 * cc + dsel;
      az[cc].x = zp[0];
      az[cc].y = zp[1];
    }

    // V B-operands for all 4 N-tiles x 3 K-chunks, hoisted for clause forming.
    const float* pV = Vtab[f] + dsel * 64 + r;
    v2f bvv[4][3];
#pragma unroll
    for (int nt = 0; nt < 4; ++nt)
#pragma unroll
      for (int cc = 0; cc < 3; ++cc) {
        bvv[nt][cc].x = pV[(4 * cc) * 64 + nt * 16];
        bvv[nt][cc].y = pV[(4 * cc + 1) * 64 + nt * 16];
      }

    // -------- context GEMM + streaming stores -------------------------------
    float* outf = out + f * fstride + base;
#pragma unroll
    for (int nt = 0; nt < 4; ++nt) {
      v8f cacc = {};
#pragma unroll
      for (int cc = 0; cc < 3; ++cc)
        cacc = wmma4(az[cc], bvv[nt][cc], cacc);

      const int col = nt * 16 + r;
      if (active) {
        // rows q = p + 8*half: p<4 valid for both halves; p>=4 only for half==0
#pragma unroll
        for (int p = 0; p < 4; ++p)
          __builtin_nontemporal_store(cacc[p],
              outf + (size_t)(p + 8 * half) * 64 + col);
        if (half == 0) {
#pragma unroll
          for (int p = 4; p < 8; ++p)
            __builtin_nontemporal_store(cacc[p],
                outf + (size_t)p * 64 + col);
        }
      }
    }
  }
}

extern "C" void kernel_launch(void* const* d_in, const int* in_sizes, int n_in,
                              void* d_out, int out_size, void* d_ws, size_t ws_size,
                              hipStream_t stream) {
  const float* Qf  = (const float*)d_in[0];
  const float* Kf  = (const float*)d_in[1];
  const float* Vf  = (const float*)d_in[2];
  const float* Qs  = (const float*)d_in[3];
  const float* Ks  = (const float*)d_in[4];
  const float* Vs  = (const float*)d_in[5];
  const float* Kj  = (const float*)d_in[6];
  const float* Vfp = (const float*)d_in[7];
  const int*   msk = (const int*)d_in[8];
  float* out = (float*)d_out;

  const int nodes  = in_sizes[0] / 768;       // B*H*L1 = 39296
  const int blocks = (nodes + 7) / 8;         // 8 waves (nodes) per 256-thread block

  fused_temporal_cross_attn<<<blocks, 256, 0, stream>>>(
      Qf, Kf, Vf, Qs, Ks, Vs, Kj, Vfp, msk, out, nodes);
}